// GraphMix_6725918785702
// MI455X (gfx1250) — compile-verified
//
#include <hip/hip_runtime.h>

#define N_NODES 100000
#define N_EDGES 3200000
#define IN_DIM  64
#define HID_DIM 32
#define N_CLASS 20

typedef float v2f __attribute__((ext_vector_type(2)));
typedef float v8f __attribute__((ext_vector_type(8)));

// ---------------- init: agg = 0, deg = 1 (self loop) ----------------
__global__ void gm_init_kernel(float* __restrict__ agg, float* __restrict__ deg) {
    int i = blockIdx.x * blockDim.x + threadIdx.x;
    if (i < N_NODES * HID_DIM) agg[i] = 0.0f;
    if (i < N_NODES)           deg[i] = 1.0f;
}

// ---------------- degree accumulation over edge targets ----------------
__global__ void gm_deg_kernel(const long long* __restrict__ ei, float* __restrict__ deg) {
    int e = blockIdx.x * blockDim.x + threadIdx.x;
    if (e < N_EDGES) {
        int c = (int)ei[(long long)N_EDGES + e];
        atomicAdd(&deg[c], 1.0f);
    }
}

// ---------------- deg -> deg^{-1/2} in place ----------------
__global__ void gm_dis_kernel(float* __restrict__ deg) {
    int i = blockIdx.x * blockDim.x + threadIdx.x;
    if (i < N_NODES) deg[i] = rsqrtf(deg[i]);
}

// ---------------- xw = x @ W1 via V_WMMA_F32_16X16X4_F32 ----------------
// One wave per 16-row tile; two 16x16 accumulators cover HID_DIM=32.
__global__ void gm_xw_wmma_kernel(const float* __restrict__ x,
                                  const float* __restrict__ W1,
                                  float* __restrict__ xw) {
    const int lane = threadIdx.x & 31;
    const int wave = (blockIdx.x * blockDim.x + threadIdx.x) >> 5;
    const int numTiles = N_NODES / 16;          // 6250, exact
    if (wave >= numTiles) return;               // whole-wave uniform exit

    const int m0    = wave * 16;
    const int m     = m0 + (lane & 15);         // A: row per lane
    const int khalf = (lane >> 4) * 2;          // lanes 16-31 hold K+2,K+3
    const int n     = lane & 15;                // B/C: column per lane

    v8f acc0 = {};
    v8f acc1 = {};
#pragma unroll
    for (int k = 0; k < IN_DIM; k += 4) {
        const int kb = k + khalf;
        v2f a;
        a.x = x[m * IN_DIM + kb];
        a.y = x[m * IN_DIM + kb + 1];
        v2f b0, b1v;
        b0.x  = W1[kb * HID_DIM + n];
        b0.y  = W1[(kb + 1) * HID_DIM + n];
        b1v.x = W1[kb * HID_DIM + n + 16];
        b1v.y = W1[(kb + 1) * HID_DIM + n + 16];
        acc0 = __builtin_amdgcn_wmma_f32_16x16x4_f32(false, a, false, b0,  (short)0, acc0, false, false);
        acc1 = __builtin_amdgcn_wmma_f32_16x16x4_f32(false, a, false, b1v, (short)0, acc1, false, false);
    }

    // C/D layout: VGPR r, lanes 0-15 -> M=r, lanes 16-31 -> M=r+8; N = lane&15
    const int mrow = m0 + (lane >> 4) * 8;
#pragma unroll
    for (int r = 0; r < 8; ++r) {
        xw[(mrow + r) * HID_DIM + n]      = acc0[r];
        xw[(mrow + r) * HID_DIM + n + 16] = acc1[r];
    }
}

// ---------------- edge scatter: one wave per edge, lane = hidden dim ----------------
__global__ void gm_scatter_kernel(const long long* __restrict__ ei,
                                  const float* __restrict__ dis,
                                  const float* __restrict__ xw,
                                  float* __restrict__ agg) {
    int t = blockIdx.x * blockDim.x + threadIdx.x;   // up to 102.4M, fits int32
    int e = t >> 5;
    if (e >= N_EDGES) return;
    int d = t & 31;
    int r = (int)ei[e];
    int c = (int)ei[(long long)N_EDGES + e];
    float nrm = dis[r] * dis[c];
    atomicAdd(&agg[c * HID_DIM + d], nrm * xw[r * HID_DIM + d]);
}

// ---------------- out = relu(agg + dis^2*xw + b1) @ Wout + bout ----------------
// Self-loop term, bias and ReLU fused into A-fragment construction.
__global__ void gm_out_wmma_kernel(const float* __restrict__ agg,
                                   const float* __restrict__ xw,
                                   const float* __restrict__ dis,
                                   const float* __restrict__ b1,
                                   const float* __restrict__ Wout,
                                   const float* __restrict__ bout,
                                   float* __restrict__ out) {
    const int lane = threadIdx.x & 31;
    const int wave = (blockIdx.x * blockDim.x + threadIdx.x) >> 5;
    const int numTiles = N_NODES / 16;
    if (wave >= numTiles) return;

    const int m0    = wave * 16;
    const int m     = m0 + (lane & 15);
    const int khalf = (lane >> 4) * 2;
    const int n     = lane & 15;
    const int n2    = n + 16;
    const float ds  = dis[m];
    const float selfnorm = ds * ds;
    const bool n2ok = (n2 < N_CLASS);

    v8f acc0 = {};
    v8f acc1 = {};
#pragma unroll
    for (int k = 0; k < HID_DIM; k += 4) {
        const int kb = k + khalf;
        float e0 = agg[m * HID_DIM + kb]     + selfnorm * xw[m * HID_DIM + kb]     + b1[kb];
        float e1 = agg[m * HID_DIM + kb + 1] + selfnorm * xw[m * HID_DIM + kb + 1] + b1[kb + 1];
        v2f a = { fmaxf(e0, 0.0f), fmaxf(e1, 0.0f) };
        v2f bA = { Wout[kb * N_CLASS + n], Wout[(kb + 1) * N_CLASS + n] };
        v2f bB = { n2ok ? Wout[kb * N_CLASS + n2]       : 0.0f,
                   n2ok ? Wout[(kb + 1) * N_CLASS + n2] : 0.0f };
        acc0 = __builtin_amdgcn_wmma_f32_16x16x4_f32(false, a, false, bA, (short)0, acc0, false, false);
        acc1 = __builtin_amdgcn_wmma_f32_16x16x4_f32(false, a, false, bB, (short)0, acc1, false, false);
    }

    const int mrow = m0 + (lane >> 4) * 8;
    const float bo0 = bout[n];
    const float bo1 = n2ok ? bout[n2] : 0.0f;
#pragma unroll
    for (int r = 0; r < 8; ++r) {
        out[(mrow + r) * N_CLASS + n] = acc0[r] + bo0;
        if (n2ok) out[(mrow + r) * N_CLASS + n2] = acc1[r] + bo1;
    }
}

extern "C" void kernel_launch(void* const* d_in, const int* in_sizes, int n_in,
                              void* d_out, int out_size, void* d_ws, size_t ws_size,
                              hipStream_t stream) {
    const float*     x    = (const float*)d_in[0];
    const long long* ei   = (const long long*)d_in[1];   // [2, E] int64
    const float*     W1   = (const float*)d_in[2];
    const float*     b1   = (const float*)d_in[3];
    const float*     Wout = (const float*)d_in[4];
    const float*     bout = (const float*)d_in[5];
    float*           out  = (float*)d_out;

    // workspace layout: xw [N*32] | agg [N*32] | deg/dis [N]   (~26 MB)
    float* xw  = (float*)d_ws;
    float* agg = xw  + (size_t)N_NODES * HID_DIM;
    float* deg = agg + (size_t)N_NODES * HID_DIM;

    const int B = 256;

    gm_init_kernel<<<(N_NODES * HID_DIM + B - 1) / B, B, 0, stream>>>(agg, deg);
    gm_deg_kernel<<<(N_EDGES + B - 1) / B, B, 0, stream>>>(ei, deg);
    gm_dis_kernel<<<(N_NODES + B - 1) / B, B, 0, stream>>>(deg);

    const int gemmThreads = (N_NODES / 16) * 32;   // one wave per 16-row tile
    gm_xw_wmma_kernel<<<(gemmThreads + B - 1) / B, B, 0, stream>>>(x, W1, xw);

    gm_scatter_kernel<<<(N_EDGES * 32 + B - 1) / B, B, 0, stream>>>(ei, deg, xw, agg);

    gm_out_wmma_kernel<<<(gemmThreads + B - 1) / B, B, 0, stream>>>(agg, xw, deg, b1, Wout, bout, out);
}